// HAMLoss_19963007992355
// MI455X (gfx1250) — compile-verified
//
#include <hip/hip_runtime.h>
#include <stdint.h>

#define VAR0  0.1f
#define VAR1  0.2f
#define K_TOP 5
#define T1C   0.35f
#define T2C   0.5f
#define ALPHA 0.25f
#define BETA  0.11f

#define BLOCK_W 256
#define BLOCK_G 512

typedef float v2f __attribute__((ext_vector_type(2)));
typedef float v8f __attribute__((ext_vector_type(8)));

// ---------------------------------------------------------------------------
// Wave32 sum via WMMA f32 16x16x4: lane L (<16) value sits at A[L][0], lane
// L>=16 at A[L-16][2] (per ISA 16x4 f32 A layout); B = ones(4x16) makes every
// D row = v_m + v_{m+16}. Per-lane sum of the 8 D VGPRs covers M=0..7 (lanes
// 0-15) or M=8..15 (lanes 16-31); one shfl_xor(16) completes the 32-lane sum.
// ---------------------------------------------------------------------------
__device__ __forceinline__ float wave_sum32(float v) {
  v2f a; a[0] = v;   a[1] = 0.0f;
  v2f b; b[0] = 1.0f; b[1] = 1.0f;
  v8f c = {};
  v8f d = __builtin_amdgcn_wmma_f32_16x16x4_f32(false, a, false, b,
                                                (short)0, c, false, false);
  float s = d[0] + d[1] + d[2] + d[3] + d[4] + d[5] + d[6] + d[7];
  s += __shfl_xor(s, 16, 32);
  return s;
}

// ---------------------------------------------------------------------------
// Async-stage `count` floats of truth data into LDS (CDNA5 async-to-LDS path).
// Must be called by all threads of the block (contains barrier).
// ---------------------------------------------------------------------------
__device__ __forceinline__ void stage_async(const float* __restrict__ src,
                                            float* lds, int count) {
  int t = threadIdx.x;
  if (t < count) {
    unsigned int laddr = (unsigned int)(uintptr_t)(lds + t);
    unsigned int voff  = (unsigned int)(t * 4);
    asm volatile("global_load_async_to_lds_b32 %0, %1, %2"
                 :: "v"(laddr), "v"(voff), "s"(src)
                 : "memory");
  }
  asm volatile("s_wait_asynccnt 0" ::: "memory");
  __syncthreads();
}

__device__ __forceinline__ float iou_box(float tx1, float ty1, float tx2, float ty2,
                                         float px1, float py1, float px2, float py2) {
  float lx = fmaxf(tx1, px1), ly = fmaxf(ty1, py1);
  float rx = fminf(tx2, px2), ry = fminf(ty2, py2);
  float w = fmaxf(rx - lx, 0.0f), h = fmaxf(ry - ly, 0.0f);
  float inter = w * h;
  float at = (tx2 - tx1) * (ty2 - ty1);
  float ap = (px2 - px1) * (py2 - py1);
  return inter / (at + ap - inter);
}

__device__ __forceinline__ void decode_box(float cx, float cy, float pw, float ph,
                                           const float4 l, float& x1, float& y1,
                                           float& x2, float& y2) {
  float dcx = cx + l.x * VAR0 * pw;
  float dcy = cy + l.y * VAR0 * ph;
  float dw  = pw * expf(l.z * VAR1);
  float dh  = ph * expf(l.w * VAR1);
  x1 = dcx - dw * 0.5f; y1 = dcy - dh * 0.5f;
  x2 = dcx + dw * 0.5f; y2 = dcy + dh * 0.5f;
}

__device__ __forceinline__ float smoothl1_sum4(const float4 l, const float* tr5,
                                               float cx, float cy, float pw, float ph) {
  float mx1 = tr5[0], my1 = tr5[1], mx2 = tr5[2], my2 = tr5[3];
  float gx = ((mx1 + mx2) * 0.5f - cx) / (VAR0 * pw);
  float gy = ((my1 + my2) * 0.5f - cy) / (VAR0 * ph);
  float gw = logf((mx2 - mx1) / pw) / VAR1;
  float gh = logf((my2 - my1) / ph) / VAR1;
  float d[4] = { l.x - gx, l.y - gy, l.z - gw, l.w - gh };
  float s = 0.0f;
  #pragma unroll
  for (int k = 0; k < 4; ++k) {
    float xv = fabsf(d[k]);
    s += (xv >= BETA) ? (xv - 0.5f * BETA) : (0.5f * xv * xv / BETA);
  }
  return s;
}

__device__ __forceinline__ float focal_elem(const float* lg, int C, float t,
                                            float fiou, bool use_fiou) {
  float keep = (t >= 0.0f) ? 1.0f : 0.0f;
  float tt = fmaxf(t, 0.0f);
  int cls = (int)tt; if (cls >= C) cls = C - 1;
  float x = lg[cls];
  float ce = fmaxf(x, 0.0f) - x * tt + log1pf(expf(-fabsf(x)));
  float a  = tt * ALPHA + (1.0f - tt) * (1.0f - ALPHA);
  float p  = 1.0f / (1.0f + expf(-x));
  float pt = (tt == 1.0f) ? p : (1.0f - p);
  if (use_fiou) a *= fiou;
  float om = 1.0f - pt;           // GAMMA == 2
  return a * om * om * ce * keep;
}

// ---------------------------------------------------------------------------
// Kernel A: per-prior best ground-truth (max/argmax over N); zero accumulators.
// ---------------------------------------------------------------------------
__global__ void k_bestprior(const float* __restrict__ priors,
                            const float* __restrict__ targets,
                            float* __restrict__ bt_score, int* __restrict__ bt_idx,
                            float* __restrict__ accum, int B, int P, int N) {
  __shared__ float s_tr[320];
  long start = (long)blockIdx.x * blockDim.x;
  long total = (long)B * P;
  int b0 = (int)(start / P);
  int b1 = (int)((start + blockDim.x - 1) / P);
  bool uni = (b0 == b1) && (b1 < B) && (N * 5 <= 320);
  stage_async(targets + (size_t)(uni ? b0 : 0) * N * 5, s_tr, uni ? N * 5 : 0);

  long gid = start + threadIdx.x;
  if (gid < total) {
    int b = (int)(gid / P), i = (int)(gid % P);
    const float* tr = uni ? s_tr : (targets + (size_t)b * N * 5);
    float4 pr = ((const float4*)priors)[i];
    float px1 = pr.x - pr.z * 0.5f, py1 = pr.y - pr.w * 0.5f;
    float px2 = pr.x + pr.z * 0.5f, py2 = pr.y + pr.w * 0.5f;
    float best = -1.0f; int bj = 0;
    for (int j = 0; j < N; ++j) {
      float v = iou_box(tr[j*5], tr[j*5+1], tr[j*5+2], tr[j*5+3], px1, py1, px2, py2);
      if (v > best) { best = v; bj = j; }
    }
    bt_score[gid] = best; bt_idx[gid] = bj;
  }
  if (gid < 8) accum[gid] = 0.0f;
}

// ---------------------------------------------------------------------------
// Kernel B: serial greedy bipartite match, one workgroup per image.
// Removed-column bitmap + row flags + argmax tree live in LDS.
// ---------------------------------------------------------------------------
__global__ void k_greedy(const float* __restrict__ priors,
                         const float* __restrict__ targets,
                         float* __restrict__ bt_score, int* __restrict__ bt_idx,
                         int P, int N) {
  extern __shared__ unsigned int smem[];
  int b = blockIdx.x, tid = threadIdx.x;
  int nb = (P + 31) >> 5;
  unsigned int* colbits = smem;
  float* s_tr = (float*)(colbits + nb);
  unsigned int* s_row = (unsigned int*)(s_tr + N * 5);
  float* rv = (float*)(s_row + 64);
  int*   rf = (int*)(rv + BLOCK_G);

  for (int k = tid; k < nb; k += BLOCK_G) colbits[k] = 0u;
  for (int k = tid; k < 64; k += BLOCK_G) s_row[k] = 0u;
  stage_async(targets + (size_t)b * N * 5, s_tr, N * 5);  // barrier inside

  for (int it = 0; it < N; ++it) {
    float bestv = -1e30f; int bestflat = 0x7fffffff;
    for (int i = tid; i < P; i += BLOCK_G) {
      if ((colbits[i >> 5] >> (i & 31)) & 1u) continue;
      float4 pr = ((const float4*)priors)[i];
      float px1 = pr.x - pr.z * 0.5f, py1 = pr.y - pr.w * 0.5f;
      float px2 = pr.x + pr.z * 0.5f, py2 = pr.y + pr.w * 0.5f;
      for (int j = 0; j < N; ++j) {
        if (s_row[j]) continue;
        float v = iou_box(s_tr[j*5], s_tr[j*5+1], s_tr[j*5+2], s_tr[j*5+3],
                          px1, py1, px2, py2);
        int flat = j * P + i;
        if (v > bestv || (v == bestv && flat < bestflat)) { bestv = v; bestflat = flat; }
      }
    }
    rv[tid] = bestv; rf[tid] = bestflat;
    __syncthreads();
    for (int s = BLOCK_G / 2; s > 0; s >>= 1) {
      if (tid < s) {
        float v2 = rv[tid + s]; int f2 = rf[tid + s];
        if (v2 > rv[tid] || (v2 == rv[tid] && f2 < rf[tid])) { rv[tid] = v2; rf[tid] = f2; }
      }
      __syncthreads();
    }
    if (tid == 0) {
      int flat = rf[0]; int j = flat / P; int i = flat % P;
      s_row[j] = 1u;
      atomicOr(&colbits[i >> 5], 1u << (i & 31));
      bt_score[(size_t)b * P + i] = rv[0];
      bt_idx[(size_t)b * P + i]   = j;
    }
    __syncthreads();
  }
}

// ---------------------------------------------------------------------------
// Kernel C: conf_s1 (pre-ignore), decoded-box best-gt max/argmax, zero c_score.
// ---------------------------------------------------------------------------
__global__ void k_stage2(const float* __restrict__ loc, const float* __restrict__ priors,
                         const float* __restrict__ targets,
                         const float* __restrict__ bt_score, const int* __restrict__ bt_idx,
                         float* __restrict__ conf1, float* __restrict__ cbp_score,
                         int* __restrict__ cbp_idx, float* __restrict__ c_score,
                         int B, int P, int N) {
  __shared__ float s_tr[320];
  long start = (long)blockIdx.x * blockDim.x;
  long total = (long)B * P;
  int b0 = (int)(start / P);
  int b1 = (int)((start + blockDim.x - 1) / P);
  bool uni = (b0 == b1) && (b1 < B) && (N * 5 <= 320);
  stage_async(targets + (size_t)(uni ? b0 : 0) * N * 5, s_tr, uni ? N * 5 : 0);

  long gid = start + threadIdx.x;
  if (gid >= total) return;
  int b = (int)(gid / P), i = (int)(gid % P);
  const float* tr = uni ? s_tr : (targets + (size_t)b * N * 5);

  conf1[gid] = (bt_score[gid] < T1C) ? 0.0f : tr[bt_idx[gid] * 5 + 4];

  float4 pr = ((const float4*)priors)[i];
  float4 lv = ((const float4*)loc)[gid];
  float dx1, dy1, dx2, dy2;
  decode_box(pr.x, pr.y, pr.z, pr.w, lv, dx1, dy1, dx2, dy2);
  float best = -1.0f; int bj = 0;
  for (int j = 0; j < N; ++j) {
    float v = iou_box(tr[j*5], tr[j*5+1], tr[j*5+2], tr[j*5+3], dx1, dy1, dx2, dy2);
    if (v > best) { best = v; bj = j; }
  }
  cbp_score[gid] = best; cbp_idx[gid] = bj; c_score[gid] = 0.0f;
}

// ---------------------------------------------------------------------------
// Kernel D: top-K_TOP per (image, target) row of thresholded c_iou.
// Sequential select-max passes reproduce lax.top_k ordering (ties -> low idx).
// ---------------------------------------------------------------------------
__global__ void k_topk(const float* __restrict__ loc, const float* __restrict__ priors,
                       const float* __restrict__ targets,
                       float* __restrict__ tk_s, int* __restrict__ tk_i,
                       int P, int N) {
  __shared__ float rv[BLOCK_W];
  __shared__ int   ri[BLOCK_W];
  __shared__ int   chosen[K_TOP];
  int b = blockIdx.x / N, j = blockIdx.x % N;
  const float* t = targets + ((size_t)b * N + j) * 5;
  float tx1 = t[0], ty1 = t[1], tx2 = t[2], ty2 = t[3];

  for (int k = 0; k < K_TOP; ++k) {
    float bestv = -1e30f; int besti = 0x7fffffff;
    for (int i = threadIdx.x; i < P; i += BLOCK_W) {
      bool skip = false;
      for (int q = 0; q < k; ++q) skip |= (chosen[q] == i);
      if (skip) continue;
      float4 pr = ((const float4*)priors)[i];
      float4 lv = ((const float4*)loc)[(size_t)b * P + i];
      float dx1, dy1, dx2, dy2;
      decode_box(pr.x, pr.y, pr.z, pr.w, lv, dx1, dy1, dx2, dy2);
      float v = iou_box(tx1, ty1, tx2, ty2, dx1, dy1, dx2, dy2);
      v = (v >= T2C) ? v : 0.0f;
      if (v > bestv || (v == bestv && i < besti)) { bestv = v; besti = i; }
    }
    rv[threadIdx.x] = bestv; ri[threadIdx.x] = besti;
    __syncthreads();
    for (int s = BLOCK_W / 2; s > 0; s >>= 1) {
      if (threadIdx.x < s) {
        float v2 = rv[threadIdx.x + s]; int i2 = ri[threadIdx.x + s];
        if (v2 > rv[threadIdx.x] || (v2 == rv[threadIdx.x] && i2 < ri[threadIdx.x])) {
          rv[threadIdx.x] = v2; ri[threadIdx.x] = i2;
        }
      }
      __syncthreads();
    }
    if (threadIdx.x == 0) {
      chosen[k] = ri[0];
      tk_s[(size_t)blockIdx.x * K_TOP + k] = rv[0];
      tk_i[(size_t)blockIdx.x * K_TOP + k] = ri[0];
    }
    __syncthreads();
  }
}

// ---------------------------------------------------------------------------
// Kernel E: serial assignment sweep (matches lax.fori_loop write order).
// ---------------------------------------------------------------------------
__global__ void k_assign(const float* __restrict__ tk_s, const int* __restrict__ tk_i,
                         const float* __restrict__ conf1,
                         int* __restrict__ cbp_idx, float* __restrict__ c_score,
                         int P, int N) {
  if (threadIdx.x != 0) return;
  int b = blockIdx.x;
  for (int t = 0; t < N * K_TOP; ++t) {
    int i = t / K_TOP, k = t % K_TOP;
    float s = tk_s[((size_t)b * N + i) * K_TOP + k];
    int   p = tk_i[((size_t)b * N + i) * K_TOP + k];
    if (conf1[(size_t)b * P + p] < 1.0f && s > 0.0f) {
      cbp_idx[(size_t)b * P + p] = i;
      c_score[(size_t)b * P + p] = s;
    }
  }
}

// ---------------------------------------------------------------------------
// Kernel F: per-element loss terms; wave reduction via WMMA; global atomics.
// accum = { n1, n2, sl1, sl2, fl1, fl2 }
// ---------------------------------------------------------------------------
__global__ void k_loss(const float* __restrict__ loc, const float* __restrict__ conf,
                       const float* __restrict__ priors, const float* __restrict__ targets,
                       const float* __restrict__ bt_score, const int* __restrict__ bt_idx,
                       const float* __restrict__ conf1, const float* __restrict__ cbp_score,
                       const int* __restrict__ cbp_idx, const float* __restrict__ c_score,
                       float* __restrict__ accum, int B, int P, int N, int C) {
  __shared__ float s_tr[320];
  __shared__ float wsums[BLOCK_W / 32][6];
  long start = (long)blockIdx.x * blockDim.x;
  long total = (long)B * P;
  int b0 = (int)(start / P);
  int b1 = (int)((start + blockDim.x - 1) / P);
  bool uni = (b0 == b1) && (b1 < B) && (N * 5 <= 320);
  stage_async(targets + (size_t)(uni ? b0 : 0) * N * 5, s_tr, uni ? N * 5 : 0);

  long gid = start + threadIdx.x;
  float vals[6] = {0.f, 0.f, 0.f, 0.f, 0.f, 0.f};
  if (gid < total) {
    int b = (int)(gid / P), i = (int)(gid % P);
    const float* tr = uni ? s_tr : (targets + (size_t)b * N * 5);
    float cs = c_score[gid];

    // final conf_s1 with ignore rule
    float c1 = conf1[gid];
    if (bt_score[gid] < T1C && cbp_score[gid] >= T2C && cs < T2C) c1 = -1.0f;
    // conf_s2
    float c2 = (cs < T2C) ? -1.0f : tr[cbp_idx[gid] * 5 + 4];

    bool m1 = c1 > 0.0f, m2 = c2 > 0.0f;
    float4 pr = ((const float4*)priors)[i];
    float4 lv = ((const float4*)loc)[gid];
    if (m1) { vals[0] = 1.0f; vals[2] = smoothl1_sum4(lv, tr + bt_idx[gid] * 5,  pr.x, pr.y, pr.z, pr.w); }
    if (m2) { vals[1] = 1.0f; vals[3] = smoothl1_sum4(lv, tr + cbp_idx[gid] * 5, pr.x, pr.y, pr.z, pr.w); }
    const float* lg = conf + gid * C;
    vals[4] = focal_elem(lg, C, c1, 0.0f, false);
    vals[5] = focal_elem(lg, C, c2, cs, true);
  }

  // Block-uniform from here on: EXEC must be all-ones for WMMA.
  int wave = threadIdx.x >> 5, lane = threadIdx.x & 31;
  #pragma unroll
  for (int c = 0; c < 6; ++c) {
    float s = wave_sum32(vals[c]);
    if (lane == 0) wsums[wave][c] = s;
  }
  __syncthreads();
  if (threadIdx.x == 0) {
    #pragma unroll
    for (int c = 0; c < 6; ++c) {
      float s = 0.0f;
      for (int w = 0; w < BLOCK_W / 32; ++w) s += wsums[w][c];
      atomicAdd(&accum[c], s);
    }
  }
}

__global__ void k_final(const float* __restrict__ accum, float* __restrict__ out) {
  if (blockIdx.x == 0 && threadIdx.x == 0) {
    float n1 = accum[0], n2 = accum[1];
    float l1 = accum[2] / fmaxf(n1, 1.0f), l2 = accum[3] / fmaxf(n2, 1.0f);
    float f1 = accum[4] / fmaxf(n1, 1.0f), f2 = accum[5] / fmaxf(n2, 1.0f);
    float loc_loss = (n1 > 0.0f ? l1 : 0.0f) + (n2 > 0.0f ? l2 : 0.0f);
    float cls_loss = (n1 > 0.0f ? f1 : 0.0f) + (n2 > 0.0f ? f2 : 0.0f);
    if (n1 == 0.0f && n2 == 0.0f) { loc_loss = 1e-4f; cls_loss = 1e-4f; }
    out[0] = loc_loss; out[1] = cls_loss;
  }
}

extern "C" void kernel_launch(void* const* d_in, const int* in_sizes, int n_in,
                              void* d_out, int out_size, void* d_ws, size_t ws_size,
                              hipStream_t stream) {
  const float* loc     = (const float*)d_in[0];   // (B,P,4)
  const float* conf    = (const float*)d_in[1];   // (B,P,C)
  const float* priors  = (const float*)d_in[2];   // (P,4)
  const float* targets = (const float*)d_in[3];   // (B,N,5)
  float* out = (float*)d_out;

  int  P = in_sizes[2] / 4;
  int  B = (int)((long)in_sizes[0] / ((long)P * 4));
  int  C = (int)((long)in_sizes[1] / ((long)B * P));
  int  N = (int)((long)in_sizes[3] / ((long)B * 5));
  size_t bp = (size_t)B * P;

  char* w = (char*)d_ws;
  float* bt_score  = (float*)w; w += bp * 4;
  int*   bt_idx    = (int*)w;   w += bp * 4;
  float* conf1     = (float*)w; w += bp * 4;
  float* cbp_score = (float*)w; w += bp * 4;
  int*   cbp_idx   = (int*)w;   w += bp * 4;
  float* c_score   = (float*)w; w += bp * 4;
  float* tk_s      = (float*)w; w += (size_t)B * N * K_TOP * 4;
  int*   tk_i      = (int*)w;   w += (size_t)B * N * K_TOP * 4;
  float* accum     = (float*)w; w += 8 * 4;
  (void)ws_size; (void)n_in; (void)out_size;

  int grid = (int)((bp + BLOCK_W - 1) / BLOCK_W);
  k_bestprior<<<grid, BLOCK_W, 0, stream>>>(priors, targets, bt_score, bt_idx, accum, B, P, N);

  size_t shg = (size_t)((P + 31) / 32) * 4 + (size_t)N * 5 * 4 + 64 * 4 + (size_t)BLOCK_G * 8;
  k_greedy<<<B, BLOCK_G, shg, stream>>>(priors, targets, bt_score, bt_idx, P, N);

  k_stage2<<<grid, BLOCK_W, 0, stream>>>(loc, priors, targets, bt_score, bt_idx,
                                         conf1, cbp_score, cbp_idx, c_score, B, P, N);
  k_topk<<<B * N, BLOCK_W, 0, stream>>>(loc, priors, targets, tk_s, tk_i, P, N);
  k_assign<<<B, 32, 0, stream>>>(tk_s, tk_i, conf1, cbp_idx, c_score, P, N);
  k_loss<<<grid, BLOCK_W, 0, stream>>>(loc, conf, priors, targets, bt_score, bt_idx,
                                       conf1, cbp_score, cbp_idx, c_score, accum, B, P, N, C);
  k_final<<<1, 32, 0, stream>>>(accum, out);
}